// LSSFPN_74345883893897
// MI455X (gfx1250) — compile-verified
//
#include <hip/hip_runtime.h>
#include <math.h>
#include <stdint.h>

typedef __attribute__((ext_vector_type(2))) float v2f;
typedef __attribute__((ext_vector_type(8))) float v8f;

#define B_      2
#define NCAM    6
#define BN      12          // B_*NCAM
#define IN_CH   512
#define OUT_CH  80
#define D_      112
#define DOUT    192         // D_ + OUT_CH
#define FH      16
#define FW      44
#define HW      704         // FH*FW
#define NX      128
#define NY      128

// ---------------- small device helpers ----------------

__device__ __forceinline__ void inv4(const float* m, float* out) {
    float inv[16];
    inv[0]  =  m[5]*m[10]*m[15] - m[5]*m[11]*m[14] - m[9]*m[6]*m[15] + m[9]*m[7]*m[14] + m[13]*m[6]*m[11] - m[13]*m[7]*m[10];
    inv[4]  = -m[4]*m[10]*m[15] + m[4]*m[11]*m[14] + m[8]*m[6]*m[15] - m[8]*m[7]*m[14] - m[12]*m[6]*m[11] + m[12]*m[7]*m[10];
    inv[8]  =  m[4]*m[9]*m[15]  - m[4]*m[11]*m[13] - m[8]*m[5]*m[15] + m[8]*m[7]*m[13] + m[12]*m[5]*m[11] - m[12]*m[7]*m[9];
    inv[12] = -m[4]*m[9]*m[14]  + m[4]*m[10]*m[13] + m[8]*m[5]*m[14] - m[8]*m[6]*m[13] - m[12]*m[5]*m[10] + m[12]*m[6]*m[9];
    inv[1]  = -m[1]*m[10]*m[15] + m[1]*m[11]*m[14] + m[9]*m[2]*m[15] - m[9]*m[3]*m[14] - m[13]*m[2]*m[11] + m[13]*m[3]*m[10];
    inv[5]  =  m[0]*m[10]*m[15] - m[0]*m[11]*m[14] - m[8]*m[2]*m[15] + m[8]*m[3]*m[14] + m[12]*m[2]*m[11] - m[12]*m[3]*m[10];
    inv[9]  = -m[0]*m[9]*m[15]  + m[0]*m[11]*m[13] + m[8]*m[1]*m[15] - m[8]*m[3]*m[13] - m[12]*m[1]*m[11] + m[12]*m[3]*m[9];
    inv[13] =  m[0]*m[9]*m[14]  - m[0]*m[10]*m[13] - m[8]*m[1]*m[14] + m[8]*m[2]*m[13] + m[12]*m[1]*m[10] - m[12]*m[2]*m[9];
    inv[2]  =  m[1]*m[6]*m[15]  - m[1]*m[7]*m[14]  - m[5]*m[2]*m[15] + m[5]*m[3]*m[14] + m[13]*m[2]*m[7]  - m[13]*m[3]*m[6];
    inv[6]  = -m[0]*m[6]*m[15]  + m[0]*m[7]*m[14]  + m[4]*m[2]*m[15] - m[4]*m[3]*m[14] - m[12]*m[2]*m[7]  + m[12]*m[3]*m[6];
    inv[10] =  m[0]*m[5]*m[15]  - m[0]*m[7]*m[13]  - m[4]*m[1]*m[15] + m[4]*m[3]*m[13] + m[12]*m[1]*m[7]  - m[12]*m[3]*m[5];
    inv[14] = -m[0]*m[5]*m[14]  + m[0]*m[6]*m[13]  + m[4]*m[1]*m[14] - m[4]*m[2]*m[13] - m[12]*m[1]*m[6]  + m[12]*m[2]*m[5];
    inv[3]  = -m[1]*m[6]*m[11]  + m[1]*m[7]*m[10]  + m[5]*m[2]*m[11] - m[5]*m[3]*m[10] - m[9]*m[2]*m[7]   + m[9]*m[3]*m[6];
    inv[7]  =  m[0]*m[6]*m[11]  - m[0]*m[7]*m[10]  - m[4]*m[2]*m[11] + m[4]*m[3]*m[10] + m[8]*m[2]*m[7]   - m[8]*m[3]*m[6];
    inv[11] = -m[0]*m[5]*m[11]  + m[0]*m[7]*m[9]   + m[4]*m[1]*m[11] - m[4]*m[3]*m[9]  - m[8]*m[1]*m[7]   + m[8]*m[3]*m[5];
    inv[15] =  m[0]*m[5]*m[10]  - m[0]*m[6]*m[9]   - m[4]*m[1]*m[10] + m[4]*m[2]*m[9]  + m[8]*m[1]*m[6]   - m[8]*m[2]*m[5];
    float det = m[0]*inv[0] + m[1]*inv[4] + m[2]*inv[8] + m[3]*inv[12];
    float r = 1.0f / det;
    #pragma unroll
    for (int i = 0; i < 16; ++i) out[i] = inv[i] * r;
}

__device__ __forceinline__ void mm4(const float* a, const float* b, float* c) {
    #pragma unroll
    for (int i = 0; i < 4; ++i)
        #pragma unroll
        for (int j = 0; j < 4; ++j) {
            float s = 0.f;
            #pragma unroll
            for (int k = 0; k < 4; ++k) s += a[i*4+k] * b[k*4+j];
            c[i*4+j] = s;
        }
}

// ---------------- kernels ----------------

__global__ void zero_f32(float* __restrict__ p, int n) {
    int i = blockIdx.x * blockDim.x + threadIdx.x;
    if (i < n) p[i] = 0.0f;
}

// Per (b,cam): Mfull = bda[b] @ sensor2ego @ inv(intrin); Ainv = inv(ida).
__global__ void setup_mats(const float* __restrict__ s2e, const float* __restrict__ intrin,
                           const float* __restrict__ ida, const float* __restrict__ bda,
                           float* __restrict__ mats) {
    int t = threadIdx.x;
    if (t >= BN) return;
    int b = t / NCAM;
    float Iinv[16], Ainv[16], comb[16], full[16];
    inv4(intrin + t*16, Iinv);
    inv4(ida + t*16, Ainv);
    mm4(s2e + t*16, Iinv, comb);
    mm4(bda + b*16, comb, full);
    float* o = mats + t*32;
    #pragma unroll
    for (int i = 0; i < 16; ++i) { o[i] = full[i]; o[16+i] = Ainv[i]; }
}

// df[bn, o, p] = sum_c w[o,c]*feats[bn,c,p] + bias[o]
// One block per (bn, nt): async-stage the 512x16 fp32 B-tile (32 KB) into LDS,
// then 12 waves (one per 16-row M-tile) run K=512 of V_WMMA_F32_16X16X4_F32.
__global__ __launch_bounds__(384) void depth_gemm(const float* __restrict__ feats,
                                                  const float* __restrict__ w,
                                                  const float* __restrict__ bias,
                                                  float* __restrict__ df) {
    __shared__ float ldsB[IN_CH * 16];           // [k][n], 64B per row -> 16B aligned segs

    const int nt = blockIdx.x;                   // 0..43
    const int bn = blockIdx.y;                   // 0..11
    const int tid = threadIdx.x;

    // ---- async global -> LDS stage of B tile: feats[bn, 0..511, nt*16 .. +16) ----
    const float* gbase = feats + (size_t)bn * IN_CH * HW + nt * 16;
    for (int i = tid; i < IN_CH * 4; i += 384) { // 2048 x 16B transfers
        int row = i >> 2, seg = i & 3;
        const float* gp = gbase + (size_t)row * HW + seg * 4;
        unsigned laddr = (unsigned)(uintptr_t)(&ldsB[row * 16 + seg * 4]);
        asm volatile("global_load_async_to_lds_b128 %0, %1, off"
                     :: "v"(laddr), "v"(gp) : "memory");
    }
    asm volatile("s_wait_asynccnt 0x0" ::: "memory");
    __syncthreads();

    // ---- WMMA: wave wv handles M-tile wv (12 waves cover DOUT=192 rows) ----
    const int wv   = tid >> 5;                   // 0..11 == mt
    const int lane = tid & 31;
    const int half = lane >> 4;                  // K pair {0,1} vs {2,3}
    const int lrow = lane & 15;                  // A: M index / B: N index

    const float* wrow = w + (size_t)(wv * 16 + lrow) * IN_CH;

    v8f acc = {};
    for (int k0 = 0; k0 < IN_CH; k0 += 4) {
        int ka = k0 + 2 * half;
        v2f a;  a.x  = wrow[ka];               a.y  = wrow[ka + 1];
        v2f bm; bm.x = ldsB[ka * 16 + lrow];   bm.y = ldsB[(ka + 1) * 16 + lrow];
        acc = __builtin_amdgcn_wmma_f32_16x16x4_f32(
                  false, a, false, bm, (short)0, acc, false, false);
    }

    int col = nt * 16 + lrow;
    float* dfo = df + (size_t)bn * DOUT * HW;
    #pragma unroll
    for (int v = 0; v < 8; ++v) {
        int row = wv * 16 + v + 8 * half;        // C/D layout: VGPR v -> M = v + 8*half
        dfo[(size_t)row * HW + col] = acc[v] + bias[row];
    }
}

// In-place softmax over the first D_ rows of each pixel column of df.
__global__ void depth_softmax(float* __restrict__ df) {
    int i = blockIdx.x * blockDim.x + threadIdx.x;   // over BN*HW columns
    if (i >= BN * HW) return;
    int bn = i / HW, p = i % HW;
    float* col = df + (size_t)bn * DOUT * HW + p;
    float m = -INFINITY;
    for (int d = 0; d < D_; ++d) m = fmaxf(m, col[(size_t)d * HW]);
    float s = 0.0f;
    for (int d = 0; d < D_; ++d) s += __expf(col[(size_t)d * HW] - m);
    float inv = 1.0f / s;
    for (int d = 0; d < D_; ++d) col[(size_t)d * HW] = __expf(col[(size_t)d * HW] - m) * inv;
}

// Fused lift + splat, one 128-thread block per pixel (bn,p):
//  phase 1: threads 0..111 compute voxel lin index + load softmaxed depth -> LDS
//  phase 2: threads 0..79 (one per channel) run-length-compress along depth,
//           one atomicAdd per (voxel-run, channel). Flush branch is wave-uniform.
__global__ __launch_bounds__(128) void lift_splat(const float* __restrict__ df,
                                                  const float* __restrict__ mats,
                                                  float* __restrict__ out) {
    __shared__ float s_depth[D_];
    __shared__ int   s_lin[D_];

    const int pix = blockIdx.x;              // 0 .. BN*HW-1
    const int bn  = pix / HW;
    const int p   = pix % HW;
    const int h   = p / FW, wp = p % FW;
    const int b   = bn / NCAM;
    const int t   = threadIdx.x;

    const float* base = df + (size_t)bn * DOUT * HW;

    if (t < D_) {
        const int d = t;
        const float fx = (float)wp * (703.0f / 43.0f);
        const float fy = (float)h  * (255.0f / 15.0f);
        const float fd = 2.0f + 0.5f * (float)d;

        const float* Mfull = mats + bn * 32;
        const float* Ainv  = Mfull + 16;

        float p0[4];
        #pragma unroll
        for (int i = 0; i < 4; ++i)
            p0[i] = Ainv[i*4+0]*fx + Ainv[i*4+1]*fy + Ainv[i*4+2]*fd + Ainv[i*4+3];
        float q0 = p0[0]*p0[2], q1 = p0[1]*p0[2], q2 = p0[2], q3 = p0[3];

        float g[3];
        #pragma unroll
        for (int i = 0; i < 3; ++i)
            g[i] = Mfull[i*4+0]*q0 + Mfull[i*4+1]*q1 + Mfull[i*4+2]*q2 + Mfull[i*4+3]*q3;

        // (geom - (vcoord - vsize/2)) / vsize; truncate toward zero == astype(int32)
        int gx = (int)((g[0] + 51.2f) / 0.8f);
        int gy = (int)((g[1] + 51.2f) / 0.8f);
        int gz = (int)((g[2] + 5.0f)  / 8.0f);
        bool valid = (gx >= 0) & (gx < NX) & (gy >= 0) & (gy < NY) & (gz >= 0) & (gz < 1);

        s_lin[d]   = valid ? (gy * NX + gx) : -1;
        s_depth[d] = base[(size_t)d * HW + p];
    }
    __syncthreads();

    if (t >= OUT_CH) return;
    const int o = t;
    const float ctx = base[(size_t)(D_ + o) * HW + p];
    float* dstb = out + (size_t)b * OUT_CH * (NY * NX) + (size_t)o * (NY * NX);

    float acc = 0.0f;
    int   cur = -1;
    for (int d = 0; d < D_; ++d) {
        int l = s_lin[d];                    // identical across lanes -> uniform branch
        if (l != cur) {
            if (cur >= 0) atomicAdd(dstb + cur, acc * ctx);
            cur = l;
            acc = 0.0f;
        }
        acc += s_depth[d];
    }
    if (cur >= 0) atomicAdd(dstb + cur, acc * ctx);
}

// ---------------- launch ----------------

extern "C" void kernel_launch(void* const* d_in, const int* in_sizes, int n_in,
                              void* d_out, int out_size, void* d_ws, size_t ws_size,
                              hipStream_t stream) {
    const float* feats   = (const float*)d_in[0];
    const float* w_depth = (const float*)d_in[1];
    const float* b_depth = (const float*)d_in[2];
    const float* s2e     = (const float*)d_in[3];
    const float* intrin  = (const float*)d_in[4];
    const float* ida     = (const float*)d_in[5];
    const float* bda     = (const float*)d_in[6];
    float* out = (float*)d_out;
    float* ws  = (float*)d_ws;

    float* mats = ws;                 // BN*32 floats
    float* df   = ws + 384;           // BN*DOUT*HW floats (~6.5 MB)

    int outN = B_ * OUT_CH * NY * NX; // 2,621,440
    zero_f32<<<(outN + 255) / 256, 256, 0, stream>>>(out, outN);

    setup_mats<<<1, 32, 0, stream>>>(s2e, intrin, ida, bda, mats);

    dim3 ggrid(FW, BN);                         // 44 x 12 blocks, 384 threads (12 waves)
    depth_gemm<<<ggrid, 384, 0, stream>>>(feats, w_depth, b_depth, df);

    int npix = BN * HW;                         // 8448
    depth_softmax<<<(npix + 255) / 256, 256, 0, stream>>>(df);

    lift_splat<<<npix, 128, 0, stream>>>(df, mats, out);
}